// SalientPixelsBCELoss_37598143710044
// MI455X (gfx1250) — compile-verified
//
#include <hip/hip_runtime.h>

// SalientPixelsBCELoss for MI455X (gfx1250).
// B=64, C=1, H=W=256 -> N=65536, K=16384.
// loss = sum over (b,n): selected ? min(softplus(-z),100) : min(softplus(z),100)
//   z = (d0+g0)-(d1+g1); selected = s_map value in per-batch top-K (stable ties).

#define B_DIM 64
#define N_DIM 65536
#define K_SEL 16384

#define NBLK2 1024        // blocks in BCE kernel
#define WAVES_PER_BLK 8   // 256 threads = 8 wave32
#define CHUNK 64          // elements reduced per WMMA issue
#define ITERS 8           // chunks per wave: 1024*8*8*64 = 4,194,304 = B*N exactly

typedef __attribute__((ext_vector_type(2))) float v2f;
typedef __attribute__((ext_vector_type(8))) float v8f;

// ---------------------------------------------------------------------------
// Kernel 1: per-batch exact K-th largest composite key via LDS-resident
// bisection. key = (bits(s_map) << 16) | (65535 - n): descending value,
// ascending index (matches lax.top_k stable tie-break). Values in [0,1) so
// the raw float bit pattern is monotonic. 256KB LDS/WG (CDNA5: 320KB max).
// ---------------------------------------------------------------------------
__global__ __launch_bounds__(1024) void topk_threshold_kernel(
    const float* __restrict__ s_map, unsigned long long* __restrict__ thresh) {
  __shared__ unsigned int toks[N_DIM];     // 256 KB
  __shared__ unsigned int cnt;
  const float* base = s_map + (size_t)blockIdx.x * N_DIM;
  for (int i = threadIdx.x; i < N_DIM; i += 1024)
    toks[i] = __float_as_uint(base[i]);
  __syncthreads();

  // invariant: count(key >= lo) >= K, count(key >= hi) < K.
  unsigned long long lo = 0ull, hi = 1ull << 46;   // bits < 2^30 -> key < 2^46
  while (hi - lo > 1ull) {
    unsigned long long mid = (lo + hi) >> 1;
    if (threadIdx.x == 0) cnt = 0u;
    __syncthreads();
    unsigned int c = 0;
    for (int i = threadIdx.x; i < N_DIM; i += 1024) {
      unsigned long long key = ((unsigned long long)toks[i] << 16) |
                               (unsigned long long)(0xFFFFu - (unsigned int)i);
      c += (key >= mid) ? 1u : 0u;
    }
    atomicAdd(&cnt, c);
    __syncthreads();
    unsigned int tot = cnt;
    __syncthreads();                        // everyone read cnt before reuse
    if (tot >= (unsigned int)K_SEL) lo = mid; else hi = mid;
  }
  if (threadIdx.x == 0) thresh[blockIdx.x] = lo;   // lo == K-th largest key
}

// ---------------------------------------------------------------------------
// Per-element BCE term (f32, numerically stable softplus, +100 cap = the
// torch BCELoss -100 log clamp).
// ---------------------------------------------------------------------------
__device__ __forceinline__ float bce_term(const float2* __restrict__ dec2,
                                          const float2* __restrict__ gum2,
                                          const float* __restrict__ smap,
                                          const unsigned long long* __restrict__ thr,
                                          int e) {
  const int b = e >> 16;            // N = 65536
  const int n = e & 0xFFFF;
  const float2 d = dec2[e];
  const float2 g = gum2[e];
  const float z = (d.x + g.x) - (d.y + g.y);
  const unsigned int bits = __float_as_uint(smap[e]);
  const unsigned long long key =
      ((unsigned long long)bits << 16) | (unsigned long long)(0xFFFFu - (unsigned int)n);
  const bool sel = key >= thr[b];
  const float x = sel ? -z : z;
  const float sp = fmaxf(x, 0.0f) + log1pf(expf(-fabsf(x)));
  return fminf(sp, 100.0f);
}

// ---------------------------------------------------------------------------
// Kernel 2: stream the 80MB once; reduce 64 f32 terms per WMMA issue using
// V_WMMA_F32_16X16X4_F32 with B = ones (D = A*1 + C keeps full f32 accuracy
// on the matrix pipe). Deterministic fixed-order combine per block.
// ---------------------------------------------------------------------------
__global__ __launch_bounds__(256) void bce_loss_kernel(
    const float* __restrict__ dec, const float* __restrict__ smap,
    const float* __restrict__ gum, const unsigned long long* __restrict__ thr,
    float* __restrict__ partials) {
  const float2* dec2 = (const float2*)dec;
  const float2* gum2 = (const float2*)gum;
  const int lane = threadIdx.x & 31;
  const int wib  = threadIdx.x >> 5;
  const int wave = blockIdx.x * WAVES_PER_BLK + wib;

  v8f acc = {};                     // f32 accumulator tile (C/D)
  v2f onesv; onesv.x = 1.0f; onesv.y = 1.0f;   // B = 4x16 all-ones

  for (int it = 0; it < ITERS; ++it) {
    const int base = (wave * ITERS + it) * CHUNK;
    v2f a;                          // A = 16x4 f32: 2 elements/lane, coalesced
    a.x = bce_term(dec2, gum2, smap, thr, base + lane);
    a.y = bce_term(dec2, gum2, smap, thr, base + 32 + lane);
    // D[m][n] = sum_k A[m][k] + C[m][n]  -> row-sum accumulate, all-f32
    acc = __builtin_amdgcn_wmma_f32_16x16x4_f32(
        /*neg_a=*/false, a, /*neg_b=*/false, onesv,
        /*c_mod=*/(short)0, acc, /*reuse_a=*/false, /*reuse_b=*/false);
  }

  // D column n is identical for every n: total = column 0 = rows 0..7 (lane 0)
  // + rows 8..15 (lane 16), each lane holding 8 VGPRs of D.
  float partial = acc[0] + acc[1] + acc[2] + acc[3] +
                  acc[4] + acc[5] + acc[6] + acc[7];
  float wsum = __shfl(partial, 0, 32) + __shfl(partial, 16, 32);

  __shared__ float wparts[WAVES_PER_BLK];
  if (lane == 0) wparts[wib] = wsum;
  __syncthreads();
  if (threadIdx.x == 0) {
    float s = 0.0f;
    for (int w = 0; w < WAVES_PER_BLK; ++w) s += wparts[w];   // fixed order
    partials[blockIdx.x] = s;
  }
}

// ---------------------------------------------------------------------------
// Kernel 3: deterministic serial combine of 1024 block partials.
// ---------------------------------------------------------------------------
__global__ void finalize_kernel(const float* __restrict__ partials,
                                float* __restrict__ out) {
  if (threadIdx.x == 0 && blockIdx.x == 0) {
    float s = 0.0f;
    for (int i = 0; i < NBLK2; ++i) s += partials[i];
    out[0] = s;
  }
}

extern "C" void kernel_launch(void* const* d_in, const int* in_sizes, int n_in,
                              void* d_out, int out_size, void* d_ws, size_t ws_size,
                              hipStream_t stream) {
  const float* dec  = (const float*)d_in[0];   // decision_scores [B,N,2]
  const float* smap = (const float*)d_in[1];   // s_map [B,1,H,W] == [B,N]
  const float* gum  = (const float*)d_in[2];   // gumbel_noise [B,N,2]

  unsigned long long* thr = (unsigned long long*)d_ws;            // 64 * 8B
  float* partials = (float*)((char*)d_ws + 512);                  // 1024 * 4B

  topk_threshold_kernel<<<B_DIM, 1024, 0, stream>>>(smap, thr);
  bce_loss_kernel<<<NBLK2, 256, 0, stream>>>(dec, smap, gum, thr, partials);
  finalize_kernel<<<1, 64, 0, stream>>>(partials, (float*)d_out);
}